// IntraSampleNTXEntLoss_8624294331077
// MI455X (gfx1250) — compile-verified
//
#include <hip/hip_runtime.h>
#include <math.h>

// MI455X (gfx1250) — memory-bound NT-Xent loss, single pass over 256 MB.
// Gram-per-b via V_WMMA_F32_16X16X4_F32; X tile held in LDS so the variance
// pass re-reads LDS, not HBM.  HBM floor ~11us @ 23.3 TB/s.

typedef __attribute__((ext_vector_type(2))) float v2f;
typedef __attribute__((ext_vector_type(4))) float v4f;
typedef __attribute__((ext_vector_type(8))) float v8f;

#define CC     4
#define VV     2
#define BB     16384
#define DD     512
#define ROWS   16        // 2 b-samples x 8 (c,v) vectors
#define PADD   516       // LDS row stride in floats (bank-conflict-free b64 reads)
#define GPAD   17
#define WAVES  4
#define TPB    128
#define TILES  (BB / 2)          // 8192
#define NBLK   (TILES / WAVES)   // 2048
#define TEMP_INV 10.0f           // 1 / TEMPERATURE
#define EPSV     1e-8f
#define NEG_INF  (-1e30f)

// ---------------------------------------------------------------- zero ws
__global__ void ntxent_zero(float* __restrict__ ws) {
  int i = blockIdx.x * blockDim.x + threadIdx.x;
  if (i < 4 * DD + NBLK) ws[i] = 0.0f;
}

// ---------------------------------------------------------------- main pass
__global__ void __launch_bounds__(TPB)
ntxent_main(const float* __restrict__ x, float* __restrict__ ws) {
  extern __shared__ float tile[];            // [WAVES][ROWS][PADD]  (~129 KB dynamic)
  __shared__ float gbuf[WAVES][ROWS][GPAD];  // per-wave 16x16 gram dump
  __shared__ float nbuf[WAVES][ROWS];        // per-wave norms
  __shared__ float vacc[4 * DD];             // block variance accumulators
  __shared__ float cebuf;                    // block ce partial

  const int tid  = threadIdx.x;
  const int wave = tid >> 5;
  const int lane = tid & 31;

  for (int i = tid; i < 4 * DD; i += TPB) vacc[i] = 0.0f;
  if (tid == 0) cebuf = 0.0f;
  __syncthreads();

  const int t  = blockIdx.x * WAVES + wave;  // tile index
  const int b0 = t * 2;
  float* myT = tile + (size_t)wave * ROWS * PADD;

  // ---- phase 1: load 16 rows x 512 f32 into LDS (coalesced b128)
  for (int r = 0; r < ROWS; ++r) {
    const int b  = b0 + (r >> 3);
    const int cv = r & 7;                    // cv = c*2 + v  (row order)
    const float* rowp = x + ((size_t)cv * BB + b) * DD;
    #pragma unroll
    for (int cc = 0; cc < 4; ++cc) {
      const int col = (cc * 32 + lane) * 4;
      *(v4f*)(myT + r * PADD + col) = *(const v4f*)(rowp + col);
    }
  }
  __syncthreads();

  // ---- phase 2: G = X * X^T  via V_WMMA_F32_16X16X4_F32.
  // For symmetric gram, the per-lane A(16x4) and B(4x16) register images are
  // identical: lane l<16 holds X[l][k..k+1], lane l>=16 holds X[l-16][k+2..k+3].
  v8f acc = {};
  {
    const int grow  = lane & 15;
    const int ghalf = (lane >> 4) * 2;
    const float* lp = myT + grow * PADD + ghalf;
    #pragma unroll 8
    for (int k = 0; k < DD; k += 4) {
      v2f a = *(const v2f*)(lp + k);         // ds_load_b64, conflict-free (PADD=516)
      acc = __builtin_amdgcn_wmma_f32_16x16x4_f32(
          false, a, false, a, (short)0, acc, false, false);
    }
  }

  // ---- phase 3: dump G to LDS.  C/D layout: VGPR j, lane l -> M=j+8*(l>>4), N=l&15
  {
    const int Nn  = lane & 15;
    const int mhi = (lane >> 4) * 8;
    #pragma unroll
    for (int j = 0; j < 8; ++j)
      gbuf[wave][mhi + j][Nn] = acc[j];
  }
  __syncthreads();

  // norms from the gram diagonal: ||x_r|| = sqrt(G_rr)
  if (lane < ROWS) {
    float g = gbuf[wave][lane][lane];
    nbuf[wave][lane] = fmaxf(sqrtf(fmaxf(g, 0.0f)), 1e-12f);
  }
  __syncthreads();

  // ---- phase 4: cross-entropy.  lanes 0..7: (b_local, crop i)
  if (lane < 8) {
    const int bl   = lane >> 2;
    const int i    = lane & 3;
    const int base = bl * 8;
    const int ra   = base + 2 * i;                       // row of a0[i]
    const float inva = 1.0f / nbuf[wave][ra];

    const float pos = gbuf[wave][ra][ra + 1] * inva / nbuf[wave][ra + 1];
    const float l0  = pos * TEMP_INV;

    float lv[8];
    float m = l0;
    #pragma unroll
    for (int jv = 0; jv < 8; ++jv) {
      const int col = base + jv;
      float s = gbuf[wave][ra][col] * inva / nbuf[wave][col];
      float g = ((jv >> 1) == i) ? NEG_INF : s * TEMP_INV; // mask same-crop
      lv[jv] = g;
      m = fmaxf(m, g);
    }
    float ssum = expf(l0 - m);
    #pragma unroll
    for (int jv = 0; jv < 8; ++jv) ssum += expf(lv[jv] - m);
    const float ce = m + logf(ssum) - l0;
    atomicAdd(&cebuf, ce);                   // ds_add_f32, 32 adds per block
  }

  // ---- phase 5: variance statistics of normalized vectors (from LDS tile)
  {
    float invn[ROWS];
    #pragma unroll
    for (int r = 0; r < ROWS; ++r) invn[r] = 1.0f / nbuf[wave][r];
    for (int it = 0; it < DD / 32; ++it) {
      const int d = lane + 32 * it;
      float s0 = 0.f, q0 = 0.f, s1 = 0.f, q1 = 0.f;
      #pragma unroll
      for (int p = 0; p < 8; ++p) {
        float e0 = myT[(2 * p) * PADD + d] * invn[2 * p];          // v=0 rows
        float e1 = myT[(2 * p + 1) * PADD + d] * invn[2 * p + 1];  // v=1 rows
        s0 += e0; q0 += e0 * e0;
        s1 += e1; q1 += e1 * e1;
      }
      atomicAdd(&vacc[d],          s0);      // ds_add_f32, banks conflict-free
      atomicAdd(&vacc[DD + d],     q0);
      atomicAdd(&vacc[2 * DD + d], s1);
      atomicAdd(&vacc[3 * DD + d], q1);
    }
  }
  __syncthreads();

  // ---- flush block accumulators
  for (int i = tid; i < 4 * DD; i += TPB)
    atomicAdd(&ws[i], vacc[i]);              // global_atomic_add_f32
  if (tid == 0) ws[4 * DD + blockIdx.x] = cebuf;
}

// ---------------------------------------------------------------- finalize
__global__ void ntxent_final(const float* __restrict__ ws, float* __restrict__ out) {
  __shared__ float r0[512], r1[512], rc[512];
  const int tid = threadIdx.x;
  const float n = (float)(CC * BB);          // 65536 samples per dimension

  float s0 = ws[tid], q0 = ws[DD + tid];
  float s1 = ws[2 * DD + tid], q1 = ws[3 * DD + tid];
  float v0 = (q0 - s0 * s0 / n) / (n - 1.0f);
  float v1 = (q1 - s1 * s1 / n) / (n - 1.0f);
  r0[tid] = fmaxf(0.0f, 1.0f - sqrtf(v0 + EPSV));
  r1[tid] = fmaxf(0.0f, 1.0f - sqrtf(v1 + EPSV));

  float cp = 0.f;
  for (int i = tid; i < NBLK; i += 512) cp += ws[4 * DD + i];
  rc[tid] = cp;
  __syncthreads();

  for (int s = 256; s > 0; s >>= 1) {
    if (tid < s) { r0[tid] += r0[tid + s]; r1[tid] += r1[tid + s]; rc[tid] += rc[tid + s]; }
    __syncthreads();
  }
  if (tid == 0) {
    const float inv_loss = rc[0] / n;
    const float var_loss = 0.5f * (r0[0] / (float)DD + r1[0] / (float)DD);
    out[0] = 0.9f * inv_loss + 0.1f * var_loss;
  }
}

// ---------------------------------------------------------------- launcher
extern "C" void kernel_launch(void* const* d_in, const int* in_sizes, int n_in,
                              void* d_out, int out_size, void* d_ws, size_t ws_size,
                              hipStream_t stream) {
  (void)in_sizes; (void)n_in; (void)out_size; (void)ws_size;
  const float* x = (const float*)d_in[0];
  float* ws  = (float*)d_ws;
  float* out = (float*)d_out;

  ntxent_zero<<<(4 * DD + NBLK + 255) / 256, 256, 0, stream>>>(ws);

  const size_t shmem = (size_t)WAVES * ROWS * PADD * sizeof(float);  // ~129 KB
  ntxent_main<<<NBLK, TPB, shmem, stream>>>(x, ws);

  ntxent_final<<<1, 512, 0, stream>>>(ws, out);
}